// LatticeEncoder_41961830482391
// MI455X (gfx1250) — compile-verified
//
#include <hip/hip_runtime.h>
#include <hip/hip_bf16.h>

typedef __attribute__((ext_vector_type(2))) float v2f;
typedef __attribute__((ext_vector_type(4))) float v4f;
typedef __attribute__((ext_vector_type(8))) float v8f;

__device__ __forceinline__ v8f wmma_f32(v2f a, v2f b, v8f c) {
    // D = A(16x4,f32) * B(4x16,f32) + C(16x16,f32)
    return __builtin_amdgcn_wmma_f32_16x16x4_f32(
        /*neg_a=*/false, a, /*neg_b=*/false, b,
        /*c_mod=*/(short)0, c, /*reuse_a=*/false, /*reuse_b=*/false);
}

// ---------------- K0: row norms ----------------
__global__ void rownorm_kernel(const float* __restrict__ z, float* __restrict__ xx, int N) {
    int i = blockIdx.x * blockDim.x + threadIdx.x;
    if (i >= N) return;
    float s = 0.f;
    const float* zr = z + (size_t)i * 32;
#pragma unroll
    for (int d = 0; d < 32; ++d) s += zr[d] * zr[d];
    xx[i] = s;
}

// ---------------- K1: dist2 tiles via fp32 WMMA ----------------
// one wave (32 threads) computes a 16x16 tile of ||zi-zj||^2 (diag -> +inf)
__global__ void dist2_kernel(const float* __restrict__ z, const float* __restrict__ xx,
                             float* __restrict__ dist, int N) {
    const int lane = threadIdx.x;
    const int i0 = blockIdx.x * 16;
    const int j0 = blockIdx.y * 16;
    const int m  = lane & 15;
    const int kh = (lane >> 4) * 2;   // 0 or 2
    const int rr = (lane >> 4) * 8;   // row offset for C frag

    v8f acc = {};
    const float* za = z + (size_t)(i0 + m) * 32 + kh;
    const float* zb = z + (size_t)(j0 + m) * 32 + kh;
#pragma unroll
    for (int kb = 0; kb < 32; kb += 4) {
        v2f a = *(const v2f*)(za + kb);
        v2f b = *(const v2f*)(zb + kb);
        acc = wmma_f32(a, b, acc);
    }

    const float xj = xx[j0 + m];
    const int j = j0 + m;
#pragma unroll
    for (int r = 0; r < 8; ++r) {
        int i = i0 + r + rr;
        float v = xx[i] + xj - 2.0f * acc[r];
        v = (i == j) ? __builtin_inff() : fmaxf(v, 0.0f);
        __builtin_nontemporal_store(v, &dist[(size_t)i * N + j]);
    }
}

// ---------------- K2: per-row top-8 smallest ----------------
__global__ void topk_kernel(const float* __restrict__ dist, int* __restrict__ idx, int N) {
    const int row = blockIdx.x;
    const int tid = threadIdx.x;
    const float* dr = dist + (size_t)row * N;

    float bv[8]; int bi[8];
#pragma unroll
    for (int t = 0; t < 8; ++t) { bv[t] = __builtin_inff(); bi[t] = 0; }

    for (int j = tid; j < N; j += 256) {
        float v = __builtin_nontemporal_load(&dr[j]);
        if (v < bv[7]) {
            bv[7] = v; bi[7] = j;
#pragma unroll
            for (int t = 7; t > 0; --t) {
                if (bv[t] < bv[t - 1]) {
                    float tv = bv[t]; bv[t] = bv[t - 1]; bv[t - 1] = tv;
                    int   ti = bi[t]; bi[t] = bi[t - 1]; bi[t - 1] = ti;
                }
            }
        }
    }

    __shared__ float sv[256 * 8];
    __shared__ int   si[256 * 8];
#pragma unroll
    for (int t = 0; t < 8; ++t) { sv[tid * 8 + t] = bv[t]; si[tid * 8 + t] = bi[t]; }
    __syncthreads();

    for (int s = 128; s > 0; s >>= 1) {
        if (tid < s) {
            float av[8], cv[8]; int ai[8], ci[8];
#pragma unroll
            for (int t = 0; t < 8; ++t) {
                av[t] = sv[tid * 8 + t];       ai[t] = si[tid * 8 + t];
                cv[t] = sv[(tid + s) * 8 + t]; ci[t] = si[(tid + s) * 8 + t];
            }
            float mv[8]; int mi[8];
            int pa = 0, pb = 0;
            for (int t = 0; t < 8; ++t) {
                if (pb >= 8 || (pa < 8 && av[pa] <= cv[pb])) { mv[t] = av[pa]; mi[t] = ai[pa]; ++pa; }
                else                                          { mv[t] = cv[pb]; mi[t] = ci[pb]; ++pb; }
            }
#pragma unroll
            for (int t = 0; t < 8; ++t) { sv[tid * 8 + t] = mv[t]; si[tid * 8 + t] = mi[t]; }
        }
        __syncthreads();
    }
    if (tid < 8) idx[(size_t)row * 8 + tid] = si[tid];
}

// ---------------- K3a: zero the adjacency region ----------------
__global__ void zero_kernel(float* __restrict__ p, long long n4) {
    long long i = (long long)blockIdx.x * blockDim.x + threadIdx.x;
    long long stride = (long long)gridDim.x * blockDim.x;
    v4f zv = {0.f, 0.f, 0.f, 0.f};
    v4f* p4 = (v4f*)p;
    for (; i < n4; i += stride) __builtin_nontemporal_store(zv, &p4[i]);
}

// ---------------- K3b: scatter symmetric edges ----------------
__global__ void scatter_kernel(const int* __restrict__ idx, float* __restrict__ A, int N) {
    int e = blockIdx.x * blockDim.x + threadIdx.x;
    if (e >= N * 8) return;
    int i = e >> 3;
    int j = idx[e];
    A[(size_t)i * N + j] = 1.0f;   // same-value races are benign
    A[(size_t)j * N + i] = 1.0f;
}

// ---------------- K4: fc1 + ReLU ----------------
__global__ void fc1_kernel(const float* __restrict__ z, const float* __restrict__ W1,
                           const float* __restrict__ b1, float* __restrict__ h1, int N) {
    int i = blockIdx.x;
    int h = threadIdx.x;        // 64 threads
    __shared__ float zr[32];
    if (h < 32) zr[h] = z[(size_t)i * 32 + h];
    __syncthreads();
    float s = b1[h];
#pragma unroll
    for (int d = 0; d < 32; ++d) s += zr[d] * W1[d * 64 + h];
    h1[(size_t)i * 64 + h] = fmaxf(s, 0.f);
}

// ---------------- K5: mean aggregation (A@h1, deg via ones-WMMA) + fc2 + ReLU ----
// one wave computes 16 rows x 64 cols of the output
__global__ void agg_fc2_kernel(const float* __restrict__ A, const float* __restrict__ h1,
                               const float* __restrict__ W2, const float* __restrict__ b2,
                               float* __restrict__ out, int N) {
    const int lane = threadIdx.x;
    const int i0 = blockIdx.x * 16;
    const int m  = lane & 15;
    const int kh = (lane >> 4) * 2;
    const int rr = (lane >> 4) * 8;

    v8f acc0 = {}, acc1 = {}, acc2 = {}, acc3 = {}, dacc = {};
    v2f ones; ones.x = 1.f; ones.y = 1.f;

    const float* arow = A + (size_t)(i0 + m) * N + kh;
    for (int kb = 0; kb < N; kb += 4) {
        v2f a = *(const v2f*)(arow + kb);
        const float* hb0 = h1 + (size_t)(kb + kh) * 64 + m;
        const float* hb1 = hb0 + 64;
        v2f b;
        b.x = hb0[0];  b.y = hb1[0];  acc0 = wmma_f32(a, b, acc0);
        b.x = hb0[16]; b.y = hb1[16]; acc1 = wmma_f32(a, b, acc1);
        b.x = hb0[32]; b.y = hb1[32]; acc2 = wmma_f32(a, b, acc2);
        b.x = hb0[48]; b.y = hb1[48]; acc3 = wmma_f32(a, b, acc3);
        dacc = wmma_f32(a, ones, dacc);   // row degrees, same C layout
    }

    // agg = acc / (deg + eps)  -> LDS tile [16][64]
    __shared__ float tile[16 * 64];
#pragma unroll
    for (int r = 0; r < 8; ++r) {
        float invd = 1.0f / (dacc[r] + 1e-6f);
        int row = r + rr;
        tile[row * 64 +  0 + m] = acc0[r] * invd;
        tile[row * 64 + 16 + m] = acc1[r] * invd;
        tile[row * 64 + 32 + m] = acc2[r] * invd;
        tile[row * 64 + 48 + m] = acc3[r] * invd;
    }
    __syncthreads();

    // fc2: out_tile = relu(agg_tile @ W2 + b2)
    v8f o0 = {}, o1 = {}, o2 = {}, o3 = {};
#pragma unroll
    for (int kb = 0; kb < 64; kb += 4) {
        v2f a;
        a.x = tile[m * 64 + kb + kh];
        a.y = tile[m * 64 + kb + kh + 1];
        const float* w0 = W2 + (size_t)(kb + kh) * 64 + m;
        const float* w1 = w0 + 64;
        v2f b;
        b.x = w0[0];  b.y = w1[0];  o0 = wmma_f32(a, b, o0);
        b.x = w0[16]; b.y = w1[16]; o1 = wmma_f32(a, b, o1);
        b.x = w0[32]; b.y = w1[32]; o2 = wmma_f32(a, b, o2);
        b.x = w0[48]; b.y = w1[48]; o3 = wmma_f32(a, b, o3);
    }
#pragma unroll
    for (int r = 0; r < 8; ++r) {
        size_t row = (size_t)(i0 + r + rr) * 64;
        out[row +  0 + m] = fmaxf(o0[r] + b2[ 0 + m], 0.f);
        out[row + 16 + m] = fmaxf(o1[r] + b2[16 + m], 0.f);
        out[row + 32 + m] = fmaxf(o2[r] + b2[32 + m], 0.f);
        out[row + 48 + m] = fmaxf(o3[r] + b2[48 + m], 0.f);
    }
}

extern "C" void kernel_launch(void* const* d_in, const int* in_sizes, int n_in,
                              void* d_out, int out_size, void* d_ws, size_t ws_size,
                              hipStream_t stream) {
    const float* z  = (const float*)d_in[0];
    const float* W1 = (const float*)d_in[1];
    const float* b1 = (const float*)d_in[2];
    const float* W2 = (const float*)d_in[3];
    const float* b2 = (const float*)d_in[4];
    const int N = in_sizes[0] / 32;          // D = 32

    float* out  = (float*)d_out;
    float* Abuf = out + (size_t)N * 64;      // A region; also reused as dist2 scratch

    // workspace layout: xx[N] | idx[N*8] | h1[N*64]   (~3.6 MB)
    float* xx  = (float*)d_ws;
    int*   idx = (int*)(xx + N);
    float* h1  = (float*)(idx + (size_t)N * 8);

    rownorm_kernel<<<(N + 255) / 256, 256, 0, stream>>>(z, xx, N);

    dim3 gtile(N / 16, N / 16);
    dist2_kernel<<<gtile, 32, 0, stream>>>(z, xx, Abuf, N);

    topk_kernel<<<N, 256, 0, stream>>>(Abuf, idx, N);

    fc1_kernel<<<N, 64, 0, stream>>>(z, W1, b1, h1, N);

    long long n4 = (long long)N * N / 4;
    zero_kernel<<<4096, 256, 0, stream>>>(Abuf, n4);

    scatter_kernel<<<(N * 8 + 255) / 256, 256, 0, stream>>>(idx, Abuf, N);

    agg_fc2_kernel<<<N / 16, 32, 0, stream>>>(Abuf, h1, W2, b2, out, N);
}